// RIENET_47012712022645
// MI455X (gfx1250) — compile-verified
//
#include <hip/hip_runtime.h>
#include <cstdint>

// k = (pi/3.5)^2
#define KCOEF 0.80568199f

// ---------------------------------------------------------------------------
// Zero the two packed d2-bitmask grids (1 byte/voxel, 4 voxels per u32 word)
// and the scalar output. Dims are device-side scalars, so size is computed in
// the kernel and we grid-stride.
// ---------------------------------------------------------------------------
__global__ void rscc_zero(uint32_t* __restrict__ ws,
                          const int* gd, const int* gh, const int* gw,
                          float* __restrict__ out) {
    long long voxels = (long long)(*gd) * (long long)(*gh) * (long long)(*gw);
    long long words2 = 2 * (voxels >> 2);
    long long i = (long long)blockIdx.x * blockDim.x + threadIdx.x;
    if (i == 0) *out = 0.0f;
    long long stride = (long long)gridDim.x * blockDim.x;
    for (; i < words2; i += stride) ws[i] = 0u;
}

// ---------------------------------------------------------------------------
// Splat: one thread per (atom, dx, dy); walks the 7-voxel z-run, merging
// consecutive bits that land in the same 32-bit word (z is the fastest axis,
// 4 voxels/word) so each thread issues ~2 atomicOr ops. Bit d of a voxel's
// byte set <=> some atom observed squared distance d (d<=6). OR is
// order-independent and safe under 4-voxel packing.
// Matches reference semantics: floor() voxel index, validity test on the
// center index, window coords clamped to [0, min(D,H,W)-1] on ALL axes, and
// d2 computed from the clamped coordinates.
// ---------------------------------------------------------------------------
__global__ void rscc_splat(const float* __restrict__ coords, int n,
                           const float* __restrict__ origin,
                           const float* __restrict__ voxel,
                           const int* gd, const int* gh, const int* gw,
                           uint32_t* __restrict__ wsBase, int which) {
    int D = *gd, H = *gh, W = *gw;
    long long voxels = (long long)D * H * W;
    uint32_t* mask = wsBase + (long long)which * (voxels >> 2);
    int mclamp = min(D, min(H, W)) - 1;
    float vx = *voxel;
    float ox = origin[0], oy = origin[1], oz = origin[2];

    long long total  = (long long)n * 49;
    long long stride = (long long)gridDim.x * blockDim.x;
    for (long long t = (long long)blockIdx.x * blockDim.x + threadIdx.x;
         t < total; t += stride) {
        int a  = (int)(t / 49);
        int o  = (int)(t % 49);
        int dx = o / 7 - 3;
        int dy = o % 7 - 3;

        float cx = coords[a], cy = coords[n + a], cz = coords[2 * n + a];
        int ix = (int)floorf((cx - ox) / vx);
        int iy = (int)floorf((cy - oy) / vx);
        int iz = (int)floorf((cz - oz) / vx);
        if (ix < 0 || ix >= D || iy < 0 || iy >= H || iz < 0 || iz >= W) continue;

        int px = min(max(ix + dx, 0), mclamp);
        int py = min(max(iy + dy, 0), mclamp);
        int d2xy = (px - ix) * (px - ix) + (py - iy) * (py - iy);
        if (d2xy > 6) continue;

        long long rowBase = ((long long)px * H + py) * (long long)W;
        int      curWord = -1;
        uint32_t curBits = 0;
        for (int k = -3; k <= 3; ++k) {
            int pz = min(max(iz + k, 0), mclamp);
            int d2 = d2xy + (pz - iz) * (pz - iz);
            if (d2 > 6) continue;
            long long lin = rowBase + pz;
            int      w   = (int)(lin >> 2);
            uint32_t bit = (1u << d2) << ((int)(lin & 3) * 8);
            if (w == curWord) {
                curBits |= bit;
            } else {
                if (curWord >= 0) atomicOr(&mask[curWord], curBits);
                curWord = w;
                curBits = bit;
            }
        }
        if (curWord >= 0) atomicOr(&mask[curWord], curBits);
    }
}

// ---------------------------------------------------------------------------
// Reduction: Sum over voxels of s*(s - t), where s = exp(-k * min d2) decoded
// from the src bitmask byte (ctz of the byte) and t likewise from tgt.
// Both grids are streamed through LDS with the CDNA5 per-lane asynchronous
// global->LDS copy path (global_load_async_to_lds_b128, ASYNCcnt-tracked,
// completed with s_wait_asynccnt). Each lane stages 16B from each grid per
// iteration; 33.5 MB total traffic -> ~1.5 us at 23.3 TB/s HBM (L2-resident
// after the splat phase, so effectively L2 bandwidth).
// ---------------------------------------------------------------------------
__global__ void rscc_reduce(const uint32_t* __restrict__ wsBase,
                            const int* gd, const int* gh, const int* gw,
                            float* __restrict__ out) {
    long long voxels = (long long)(*gd) * (long long)(*gh) * (long long)(*gw);
    long long words  = voxels >> 2;
    const uint32_t* A = wsBase;           // src grid bitmasks
    const uint32_t* B = wsBase + words;   // tgt grid bitmasks

    __shared__ uint4 ldsA[256];
    __shared__ uint4 ldsB[256];
    __shared__ float red[256];

    int tid = threadIdx.x;
    // Flat shared-pointer low 32 bits == raw LDS byte offset on gfx1250
    // (shared aperture lives entirely in addr[63:32]).
    unsigned ldsOffA = (unsigned)(uintptr_t)(&ldsA[tid]);
    unsigned ldsOffB = (unsigned)(uintptr_t)(&ldsB[tid]);

    double acc = 0.0;
    long long stride = (long long)gridDim.x * blockDim.x * 4;
    for (long long w0 = ((long long)blockIdx.x * blockDim.x + tid) * 4;
         w0 < words; w0 += stride) {
        uint4 wa, wb;
        if (w0 + 3 < words) {
            uint64_t ga = (uint64_t)(uintptr_t)(A + w0);
            uint64_t gb = (uint64_t)(uintptr_t)(B + w0);
            asm volatile(
                "global_load_async_to_lds_b128 %0, %2, off\n\t"
                "global_load_async_to_lds_b128 %1, %3, off\n\t"
                "s_wait_asynccnt 0"
                :
                : "v"(ldsOffA), "v"(ldsOffB), "v"(ga), "v"(gb)
                : "memory");
            wa = ldsA[tid];
            wb = ldsB[tid];
        } else {  // tail (never taken for 256^3, kept for generality)
            wa = make_uint4(0, 0, 0, 0);
            wb = make_uint4(0, 0, 0, 0);
            for (int q = 0; q < 4 && w0 + q < words; ++q) {
                ((uint32_t*)&wa)[q] = A[w0 + q];
                ((uint32_t*)&wb)[q] = B[w0 + q];
            }
        }

        const uint32_t* pa = (const uint32_t*)&wa;
        const uint32_t* pb = (const uint32_t*)&wb;
#pragma unroll
        for (int q = 0; q < 4; ++q) {
            uint32_t va = pa[q];
            if (va == 0u) continue;   // src byte 0 => contribution 0 (common)
            uint32_t vb = pb[q];
#pragma unroll
            for (int b = 0; b < 4; ++b) {
                uint32_t ba = (va >> (8 * b)) & 0xFFu;
                if (ba == 0u) continue;
                uint32_t bb = (vb >> (8 * b)) & 0xFFu;
                float s = __expf(-KCOEF * (float)__builtin_ctz(ba));
                float t = bb ? __expf(-KCOEF * (float)__builtin_ctz(bb)) : 0.0f;
                acc += (double)(s * (s - t));
            }
        }
    }

    red[tid] = (float)acc;
    __syncthreads();
    for (int s = blockDim.x >> 1; s > 0; s >>= 1) {
        if (tid < s) red[tid] += red[tid + s];
        __syncthreads();
    }
    if (tid == 0) atomicAdd(out, red[0]);
}

// ---------------------------------------------------------------------------
extern "C" void kernel_launch(void* const* d_in, const int* in_sizes, int n_in,
                              void* d_out, int out_size, void* d_ws, size_t ws_size,
                              hipStream_t stream) {
    const float* src    = (const float*)d_in[0];  // (1,3,N) flat
    const float* tgt    = (const float*)d_in[1];  // (1,3,N) flat
    const float* origin = (const float*)d_in[2];  // (3,)
    const float* voxel  = (const float*)d_in[3];  // scalar
    const int*   gd     = (const int*)d_in[4];
    const int*   gh     = (const int*)d_in[5];
    const int*   gw     = (const int*)d_in[6];
    float*       out    = (float*)d_out;
    uint32_t*    ws     = (uint32_t*)d_ws;

    int n = in_sizes[0] / 3;

    // 1) zero both bitmask grids + output (grids: 2 * D*H*W bytes in ws)
    rscc_zero<<<4096, 256, 0, stream>>>(ws, gd, gh, gw, out);

    // 2) splat src -> grid 0, tgt -> grid 1
    long long splatThreads = (long long)n * 49;
    int splatBlocks = (int)((splatThreads + 255) / 256);
    if (splatBlocks < 1) splatBlocks = 1;
    rscc_splat<<<splatBlocks, 256, 0, stream>>>(src, n, origin, voxel, gd, gh, gw, ws, 0);
    rscc_splat<<<splatBlocks, 256, 0, stream>>>(tgt, n, origin, voxel, gd, gh, gw, ws, 1);

    // 3) streamed reduction (async global->LDS staging), rho_mo - rho_em
    rscc_reduce<<<1024, 256, 0, stream>>>(ws, gd, gh, gw, out);
}